// TemporalGNN_16398185136407
// MI455X (gfx1250) — compile-verified
//
#include <hip/hip_runtime.h>
#include <math.h>

#define N_NODES 100000
#define N_EDGES 1600000
#define F_IN    128
#define OUT     32
#define PERIODS 12
#define NTILES  (N_NODES / 16)   // 6250, exact

typedef __attribute__((ext_vector_type(2))) float v2f;
typedef __attribute__((ext_vector_type(8))) float v8f;

__device__ __forceinline__ v8f wmma4(v2f a, v2f b, v8f c) {
    // D = A(16x4 f32) * B(4x16 f32) + C(16x16 f32)
    return __builtin_amdgcn_wmma_f32_16x16x4_f32(
        /*neg_a=*/false, a, /*neg_b=*/false, b,
        /*c_mod=*/(short)0, c, /*reuse_a=*/false, /*reuse_b=*/false);
}

__device__ __forceinline__ void atomAddF(float* p, float v) {
    __hip_atomic_fetch_add(p, v, __ATOMIC_RELAXED, __HIP_MEMORY_SCOPE_AGENT);
}

// ---------------- setup kernels ----------------

// deg init (self loop -> 1.0) and zero the temporal accumulator
__global__ void k_init(float* __restrict__ deg, float* __restrict__ accum) {
    int gid = blockIdx.x * blockDim.x + threadIdx.x;
    if (gid < N_NODES * OUT) accum[gid] = 0.0f;
    if (gid < N_NODES)       deg[gid]   = 1.0f;
}

__global__ void k_deg(const int* __restrict__ col, float* __restrict__ deg) {
    int gid = blockIdx.x * blockDim.x + threadIdx.x;
    if (gid < N_EDGES) atomAddF(&deg[col[gid]], 1.0f);
}

__global__ void k_rsqrt(float* __restrict__ dis) {
    int gid = blockIdx.x * blockDim.x + threadIdx.x;
    if (gid < N_NODES) {
        float d = dis[gid];
        dis[gid] = (d > 0.0f) ? rsqrtf(d) : 0.0f;
    }
}

// softmax(att) + fold conv bias through the cell GEMM: biasZ = bz@Lz1 + lbz, etc.
__global__ void k_prep(const float* __restrict__ att,
                       const float* __restrict__ bz, const float* __restrict__ Lz,
                       const float* __restrict__ lbz,
                       const float* __restrict__ bh, const float* __restrict__ Lh,
                       const float* __restrict__ lbh,
                       float* __restrict__ probs, float* __restrict__ biasZ,
                       float* __restrict__ biasH) {
    int t = threadIdx.x;
    if (t == 0) {
        float m = -1e30f;
        for (int i = 0; i < PERIODS; ++i) m = fmaxf(m, att[i]);
        float e[PERIODS]; float s = 0.0f;
        for (int i = 0; i < PERIODS; ++i) { e[i] = __expf(att[i] - m); s += e[i]; }
        float inv = 1.0f / s;
        for (int i = 0; i < PERIODS; ++i) probs[i] = e[i] * inv;
    }
    if (t < OUT) {
        float s = lbz[t];
        for (int k = 0; k < OUT; ++k) s += bz[k] * Lz[k * OUT + t];
        biasZ[t] = s;
    } else if (t < 2 * OUT) {
        int c = t - OUT;
        float s = lbh[c];
        for (int k = 0; k < OUT; ++k) s += bh[k] * Lh[k * OUT + c];
        biasH[c] = s;
    }
}

// ---------------- fused all-period feature GEMM ----------------
// xw_all[p][n][0:32]=x_p@Wz , [32:64]=x_p@Wh. One block = 16 nodes (one M tile),
// 4 waves x 3 periods. x slab for the tile is consumed with full cache-line
// utilization. B fragments are stored fragment-swizzled in LDS so each lane's
// (K,K+1) pair is one contiguous ds_load_b64; k-step outer loop reuses each B
// fragment across 3 periods.
__global__ __launch_bounds__(128) void k_xw_all(const float* __restrict__ x,
                                                const float* __restrict__ Wz,
                                                const float* __restrict__ Wh,
                                                float* __restrict__ xwall) {
    __shared__ float Wlds[F_IN * 64];  // 32 KB, swizzled: [ks][hi][j][ln][t]
    for (int i = threadIdx.x; i < F_IN * 64; i += 128) {
        int t  = i & 1;
        int ln2 = (i >> 1) & 15;
        int j  = (i >> 5) & 3;
        int h2 = (i >> 7) & 1;
        int ks = i >> 8;
        int k = ks * 4 + h2 * 2 + t;
        int c = j * 16 + ln2;
        Wlds[i] = (c < OUT) ? Wz[k * OUT + c] : Wh[k * OUT + (c - OUT)];
    }
    __syncthreads();

    const int tile = blockIdx.x;               // < NTILES (exact)
    const int nodeBase = tile * 16;
    const int wave = threadIdx.x >> 5;
    const int lane = threadIdx.x & 31;
    const int ln = lane & 15;
    const int hi = lane >> 4;                  // half-wave select
    const float* xn = x + (size_t)(nodeBase + ln) * (F_IN * PERIODS);
    const v2f* Wf = (const v2f*)Wlds;          // fragment idx = ks*128 + hi*64 + j*16 + ln
    const int fb = (hi << 6) | ln;

    v8f acc[3][4] = {};
    for (int ks = 0; ks < F_IN / 4; ++ks) {
        v2f b0 = Wf[ks * 128 + fb];
        v2f b1 = Wf[ks * 128 + fb + 16];
        v2f b2 = Wf[ks * 128 + fb + 32];
        v2f b3 = Wf[ks * 128 + fb + 48];
        const int kk0 = ks * 4 + (hi << 1);
        v2f a[3];
        #pragma unroll
        for (int pi = 0; pi < 3; ++pi) {
            const int p = wave + pi * 4;
            a[pi].x = xn[kk0 * PERIODS + p];
            a[pi].y = xn[(kk0 + 1) * PERIODS + p];
        }
        #pragma unroll
        for (int pi = 0; pi < 3; ++pi) {
            acc[pi][0] = wmma4(a[pi], b0, acc[pi][0]);
            acc[pi][1] = wmma4(a[pi], b1, acc[pi][1]);
            acc[pi][2] = wmma4(a[pi], b2, acc[pi][2]);
            acc[pi][3] = wmma4(a[pi], b3, acc[pi][3]);
        }
    }

    #pragma unroll
    for (int pi = 0; pi < 3; ++pi) {
        const int p = wave + pi * 4;
        float* outp = xwall + (size_t)p * N_NODES * 64;
        #pragma unroll
        for (int v = 0; v < 8; ++v) {
            const int m = nodeBase + (hi ? v + 8 : v);
            float* rowp = outp + (size_t)m * 64 + ln;
            rowp[0]  = acc[pi][0][v];
            rowp[16] = acc[pi][1][v];
            rowp[32] = acc[pi][2][v];
            rowp[48] = acc[pi][3][v];
        }
    }
}

// ---------------- per-period aggregation ----------------

// self-loop contribution initializes agg (avoids a zero pass)
__global__ void k_selfinit(const float* __restrict__ xw, const float* __restrict__ dis,
                           float* __restrict__ agg) {
    int gid = blockIdx.x * blockDim.x + threadIdx.x;   // N_NODES*16 threads
    if (gid >= N_NODES * 16) return;
    int n = gid >> 4, c4 = gid & 15;
    float d = dis[n]; float w = d * d;
    const float4* src = (const float4*)(xw + (size_t)n * 64) + c4;
    float4 v = *src;
    float4* dst = (float4*)(agg + (size_t)n * 64) + c4;
    float4 o; o.x = v.x * w; o.y = v.y * w; o.z = v.z * w; o.w = v.w * w;
    *dst = o;
}

// gather at source, f32 atomic scatter at target. xw/agg live in L2 (25.6 MB each).
__global__ void k_edges(const int* __restrict__ row, const int* __restrict__ col,
                        const float* __restrict__ dis, const float* __restrict__ xw,
                        float* __restrict__ agg) {
    long long gid = (long long)blockIdx.x * blockDim.x + threadIdx.x;
    if (gid >= (long long)N_EDGES * 16) return;
    int e = (int)(gid >> 4), c4 = (int)(gid & 15);
    int r = row[e], c = col[e];
    float w = dis[r] * dis[c];
    const float4* src = (const float4*)(xw + (size_t)r * 64) + c4;
    float4 v = *src;
    float* dst = agg + (size_t)c * 64 + (c4 << 2);
    atomAddF(dst + 0, v.x * w);
    atomAddF(dst + 1, v.y * w);
    atomAddF(dst + 2, v.z * w);
    atomAddF(dst + 3, v.w * w);
}

// ---------------- TGCN cell (WMMA 32x32 GEMMs + activations) ----------------
__global__ __launch_bounds__(128) void k_cell(const float* __restrict__ agg,
                                              const float* __restrict__ Lz,
                                              const float* __restrict__ Lh,
                                              const float* __restrict__ biasZ,
                                              const float* __restrict__ biasH,
                                              const float* __restrict__ probs,
                                              float* __restrict__ accum, int p) {
    __shared__ float Llds[2 * OUT * OUT];  // 8 KB, fragment-swizzled: Z then H
    for (int i = threadIdx.x; i < OUT * OUT; i += 128) {
        int t  = i & 1;
        int ln2 = (i >> 1) & 15;
        int j  = (i >> 5) & 1;
        int h2 = (i >> 6) & 1;
        int ks = i >> 7;
        int k = ks * 4 + h2 * 2 + t;
        int c = j * 16 + ln2;
        Llds[i]             = Lz[k * OUT + c];   // rows 0..31 of 64x32 Lz
        Llds[OUT * OUT + i] = Lh[k * OUT + c];
    }
    __syncthreads();

    const int wave = threadIdx.x >> 5;
    const int lane = threadIdx.x & 31;
    const int ln = lane & 15;
    const int hi = lane >> 4;
    const int tile = blockIdx.x * 4 + wave;      // wave-uniform guard: EXEC stays full
    if (tile >= NTILES) return;
    const int nodeBase = tile * 16;
    const v2f* avn = (const v2f*)(agg + (size_t)(nodeBase + ln) * 64);
    const v2f* Lf = (const v2f*)Llds;            // Z frag idx = ks*64 + hi*32 + j*16 + ln
    const int fb = (hi << 5) | ln;

    v8f z0 = {}, z1 = {}, h0 = {}, h1 = {};
    for (int ks = 0; ks < OUT / 4; ++ks) {
        const int ap0 = ks * 2 + hi;             // v2f index of (kk0, kk0+1)
        v2f az = avn[ap0];
        v2f ah = avn[16 + ap0];
        v2f bz0 = Lf[ks * 64 + fb];
        v2f bz1 = Lf[ks * 64 + fb + 16];
        v2f bh0 = Lf[512 + ks * 64 + fb];
        v2f bh1 = Lf[512 + ks * 64 + fb + 16];
        z0 = wmma4(az, bz0, z0);
        z1 = wmma4(az, bz1, z1);
        h0 = wmma4(ah, bh0, h0);
        h1 = wmma4(ah, bh1, h1);
    }

    const float pr = probs[p];
    const float bZ0 = biasZ[ln], bZ1 = biasZ[16 + ln];
    const float bH0 = biasH[ln], bH1 = biasH[16 + ln];
    #pragma unroll
    for (int v = 0; v < 8; ++v) {
        const int m = nodeBase + (hi ? v + 8 : v);
        float zz0 = 1.0f / (1.0f + __expf(-(z0[v] + bZ0)));
        float zz1 = 1.0f / (1.0f + __expf(-(z1[v] + bZ1)));
        float th0 = tanhf(h0[v] + bH0);
        float th1 = tanhf(h1[v] + bH1);
        float* ap = accum + (size_t)m * OUT;
        ap[ln]      += pr * (1.0f - zz0) * th0;  // each (m,c) owned by one lane: no race
        ap[16 + ln] += pr * (1.0f - zz1) * th1;
    }
}

// ---------------- final projection ----------------
__global__ __launch_bounds__(256) void k_out(const float* __restrict__ accum,
                                             const float* __restrict__ Wlin,
                                             const float* __restrict__ blin,
                                             float* __restrict__ out) {
    __shared__ float wl[OUT * PERIODS + PERIODS];
    for (int i = threadIdx.x; i < OUT * PERIODS; i += 256) wl[i] = Wlin[i];
    if (threadIdx.x < PERIODS) wl[OUT * PERIODS + threadIdx.x] = blin[threadIdx.x];
    __syncthreads();
    int n = blockIdx.x * 256 + threadIdx.x;
    if (n >= N_NODES) return;
    float h[OUT];
    const float* ap = accum + (size_t)n * OUT;
    #pragma unroll
    for (int o = 0; o < OUT; ++o) h[o] = fmaxf(ap[o], 0.0f);
    float* op = out + (size_t)n * PERIODS;
    #pragma unroll
    for (int q = 0; q < PERIODS; ++q) {
        float s = wl[OUT * PERIODS + q];
        #pragma unroll
        for (int o = 0; o < OUT; ++o) s += h[o] * wl[o * PERIODS + q];
        op[q] = s;
    }
}

// ---------------- host orchestration ----------------
extern "C" void kernel_launch(void* const* d_in, const int* in_sizes, int n_in,
                              void* d_out, int out_size, void* d_ws, size_t ws_size,
                              hipStream_t stream) {
    const float* x    = (const float*)d_in[0];
    const int*   ei   = (const int*)d_in[1];        // [2, E]: row then col (int32)
    const float* Wz   = (const float*)d_in[2];
    const float* bz   = (const float*)d_in[3];
    const float* Lz   = (const float*)d_in[4];
    const float* lbz  = (const float*)d_in[5];
    const float* Wh   = (const float*)d_in[6];
    const float* bh   = (const float*)d_in[7];
    const float* Lh   = (const float*)d_in[8];
    const float* lbh  = (const float*)d_in[9];
    const float* att  = (const float*)d_in[10];
    const float* Wlin = (const float*)d_in[11];
    const float* blin = (const float*)d_in[12];
    float* out = (float*)d_out;

    // workspace layout (floats), all 64B aligned
    float* ws    = (float*)d_ws;
    float* dis   = ws;                            // N (deg -> dis in place)
    float* probs = ws + 100352;                   // 12 (padded to 32)
    float* biasZ = ws + 100384;                   // 32
    float* biasH = ws + 100416;                   // 32
    float* accum = ws + 100448;                   // N*32
    float* agg   = accum + (size_t)N_NODES * OUT; // N*64
    float* xwall = agg + (size_t)N_NODES * 64;    // 12*N*64 (307 MB)

    const int* row = ei;
    const int* col = ei + N_EDGES;

    k_init<<<(N_NODES * OUT + 255) / 256, 256, 0, stream>>>(dis, accum);
    k_deg<<<(N_EDGES + 255) / 256, 256, 0, stream>>>(col, dis);
    k_rsqrt<<<(N_NODES + 255) / 256, 256, 0, stream>>>(dis);
    k_prep<<<1, 64, 0, stream>>>(att, bz, Lz, lbz, bh, Lh, lbh, probs, biasZ, biasH);
    k_xw_all<<<NTILES, 128, 0, stream>>>(x, Wz, Wh, xwall);

    for (int p = 0; p < PERIODS; ++p) {
        const float* xwp = xwall + (size_t)p * N_NODES * 64;
        k_selfinit<<<(N_NODES * 16 + 255) / 256, 256, 0, stream>>>(xwp, dis, agg);
        k_edges<<<(int)(((long long)N_EDGES * 16 + 255) / 256), 256, 0, stream>>>(row, col, dis, xwp, agg);
        k_cell<<<(NTILES + 3) / 4, 128, 0, stream>>>(agg, Lz, Lh, biasZ, biasH, probs, accum, p);
    }

    k_out<<<(N_NODES + 255) / 256, 256, 0, stream>>>(accum, Wlin, blin, out);
}